// TSControllerCoordinateTransform_75067438399925
// MI455X (gfx1250) — compile-verified
//
#include <hip/hip_runtime.h>
#include <math.h>

#define T_LEN 4000000

typedef __attribute__((ext_vector_type(2))) float v2f;
typedef __attribute__((ext_vector_type(4))) float v4f;
typedef __attribute__((ext_vector_type(8))) float v8f;

#if defined(__HIP_DEVICE_COMPILE__) && !__has_builtin(__builtin_amdgcn_wmma_f32_16x16x4_f32)
#error "__builtin_amdgcn_wmma_f32_16x16x4_f32 not available on this toolchain (device)"
#endif

// ---------------------------------------------------------------------------
// ws layout (floats): [0..11] moment accumulators (M[j*3+i]=sum xi*xj for j<3,
//                              [9..11]=sum xi), [16..24] basis rows, [25..27] refPos
// ---------------------------------------------------------------------------

__global__ void zero_ws_kernel(float* __restrict__ ws) {
  if (threadIdx.x < 16) ws[threadIdx.x] = 0.0f;
}

// Pass 1: accumulate second moments + sums of rows 0..2 via V_WMMA_F32_16X16X4_F32.
// A tile: rows 0-2 = x samples, rest 0.  B tile: dual of A with column N=3 == 1.0,
// so C[i][j] = sum xi*xj (i,j<3) and C[i][3] = sum xi.
__global__ __launch_bounds__(256) void moments_kernel(const float* __restrict__ X,
                                                      float* __restrict__ accum) {
  const int tid    = threadIdx.x;
  const int lane   = tid & 31;
  const int wave   = tid >> 5;
  const int waveId = blockIdx.x * 8 + wave;
  const int NW     = gridDim.x * 8;

  const int  M       = lane & 15;       // A-row / B-col index for this lane
  const int  hi      = lane >> 4;       // 0 -> K slots {0,1}, 1 -> K slots {2,3}
  const bool active  = (M < 3);
  const bool onesCol = (M == 3);        // B column 3 = all ones (sums)
  const float* rowp  = X + (size_t)(active ? M : 0) * (size_t)T_LEN;

  v8f ca = {0.f, 0.f, 0.f, 0.f, 0.f, 0.f, 0.f, 0.f};
  v8f cb = {0.f, 0.f, 0.f, 0.f, 0.f, 0.f, 0.f, 0.f};
  const v2f one2 = {1.0f, 1.0f};

  const int NG = T_LEN / 8;             // 8 time samples per group (2 WMMAs)
  for (int g = waveId; g < NG; g += NW) {
    const int base = g * 8 + hi * 4;    // lo half covers t..t+3, hi half t+4..t+7
    v4f f = {0.f, 0.f, 0.f, 0.f};
    if (active) f = *(const v4f*)(rowp + base);

    v2f a0 = {f.x, f.y};                // K slots (kbase, kbase+1) <- t base+{0,1}/{4,5}
    v2f a1 = {f.z, f.w};                //                          <- t base+{2,3}/{6,7}
    v2f b0 = a0, b1 = a1;               // B is the dual layout of A (symmetric product)
    if (onesCol) { b0 = one2; b1 = one2; }

#if defined(__HIP_DEVICE_COMPILE__)
    ca = __builtin_amdgcn_wmma_f32_16x16x4_f32(false, a0, false, b0, (short)0, ca, false, false);
    cb = __builtin_amdgcn_wmma_f32_16x16x4_f32(false, a1, false, b1, (short)0, cb, false, false);
#endif
  }
  v8f c = ca + cb;

  // C layout: c[i] holds row i, column = lane.  Lanes 0..3 carry moments/sums.
  __shared__ float red[12][8];
  if (lane < 4) {
    red[lane * 3 + 0][wave] = c[0];
    red[lane * 3 + 1][wave] = c[1];
    red[lane * 3 + 2][wave] = c[2];
  }
  __syncthreads();
  if (tid < 12) {
    float s = 0.f;
#pragma unroll
    for (int w = 0; w < 8; ++w) s += red[tid][w];
    atomicAdd(&accum[tid], s);
  }
}

// Tiny single-thread solve: cov -> Jacobi eig -> top eigenvector, quaternion
// sign fix, cross products -> basis + refPos into ws[16..27].
__global__ void solve_kernel(const float* __restrict__ X,
                             const float* __restrict__ refAxis,
                             float* __restrict__ ws) {
  if (threadIdx.x != 0 || blockIdx.x != 0) return;
  const float* acc = ws;
  float* br = ws + 16;
  const float Tf = (float)T_LEN;

  float sv[3] = {acc[9], acc[10], acc[11]};
  float A[3][3];
  for (int i = 0; i < 3; ++i)
    for (int j = 0; j < 3; ++j)
      A[i][j] = (acc[j * 3 + i] - sv[i] * sv[j] / Tf) / (Tf - 1.0f);

  // Jacobi eigen-decomposition of symmetric 3x3
  float V[3][3] = {{1.f,0.f,0.f},{0.f,1.f,0.f},{0.f,0.f,1.f}};
  for (int sweep = 0; sweep < 30; ++sweep) {
    float off = fabsf(A[0][1]) + fabsf(A[0][2]) + fabsf(A[1][2]);
    if (off == 0.0f) break;
    for (int k = 0; k < 3; ++k) {
      const int p = (k == 2) ? 1 : 0;
      const int q = (k == 0) ? 1 : 2;
      float apq = A[p][q];
      if (fabsf(apq) <= 1e-30f) continue;
      float theta = (A[q][q] - A[p][p]) / (2.0f * apq);
      float t  = copysignf(1.0f, theta) / (fabsf(theta) + sqrtf(theta * theta + 1.0f));
      float cc = 1.0f / sqrtf(t * t + 1.0f);
      float ssn = t * cc;
      for (int i = 0; i < 3; ++i) {            // A = A * J
        float aip = A[i][p], aiq = A[i][q];
        A[i][p] = cc * aip - ssn * aiq;
        A[i][q] = ssn * aip + cc * aiq;
      }
      for (int i = 0; i < 3; ++i) {            // A = J^T * A
        float api = A[p][i], aqi = A[q][i];
        A[p][i] = cc * api - ssn * aqi;
        A[q][i] = ssn * api + cc * aqi;
      }
      for (int i = 0; i < 3; ++i) {            // V = V * J
        float vip = V[i][p], viq = V[i][q];
        V[i][p] = cc * vip - ssn * viq;
        V[i][q] = ssn * vip + cc * viq;
      }
    }
  }
  int m = 0;
  if (A[1][1] > A[m][m]) m = 1;
  if (A[2][2] > A[m][m]) m = 2;
  float pz[3] = {V[0][m], V[1][m], V[2][m]};
  float nrm = sqrtf(pz[0]*pz[0] + pz[1]*pz[1] + pz[2]*pz[2]);
  pz[0] /= nrm; pz[1] /= nrm; pz[2] /= nrm;

  // ZFwd = quat(x[3..6][0]) applied to (0,0,1)
  const size_t T = T_LEN;
  float qx = X[3*T], qy = X[4*T], qz = X[5*T], qw = X[6*T];
  float zf0 = 2.f*qx*qz + 2.f*qw*qy;
  float zf1 = 2.f*qy*qz - 2.f*qw*qx;
  float zf2 = 1.f - 2.f*qx*qx - 2.f*qy*qy;
  if (zf0*pz[0] + zf1*pz[1] + zf2*pz[2] < 0.f) { pz[0]=-pz[0]; pz[1]=-pz[1]; pz[2]=-pz[2]; }

  float up0 = refAxis[0], up1 = refAxis[1], up2 = refAxis[2];
  float r0 = up1*pz[2] - up2*pz[1];    // right = cross(up, pz)
  float r1 = up2*pz[0] - up0*pz[2];
  float r2 = up0*pz[1] - up1*pz[0];
  float f0 = r1*up2 - r2*up1;          // fwd = cross(right, up)
  float f1 = r2*up0 - r0*up2;
  float f2 = r0*up1 - r1*up0;

  br[0]=r0;  br[1]=r1;  br[2]=r2;
  br[3]=up0; br[4]=up1; br[5]=up2;
  br[6]=f0;  br[7]=f1;  br[8]=f2;
  br[9]  = X[0];
  br[10] = X[T];
  br[11] = X[2*T];
}

// Pass 2: streaming transform + passthrough, float4 coalesced, NT stores.
__global__ __launch_bounds__(256) void transform_kernel(const float* __restrict__ X,
                                                        const float* __restrict__ br,
                                                        float* __restrict__ out) {
  const int n4 = T_LEN / 4;
  const int i = blockIdx.x * blockDim.x + threadIdx.x;
  if (i >= n4) return;

  const float b00 = br[0], b01 = br[1], b02 = br[2];
  const float b10 = br[3], b11 = br[4], b12 = br[5];
  const float b20 = br[6], b21 = br[7], b22 = br[8];
  const float r0 = br[9], r1 = br[10], r2 = br[11];

  const size_t T = T_LEN;
  const size_t o = (size_t)i * 4;

  v4f x0 = *(const v4f*)(X + 0*T + o) - r0;
  v4f x1 = *(const v4f*)(X + 1*T + o) - r1;
  v4f x2 = *(const v4f*)(X + 2*T + o) - r2;

  v4f y0 = b00*x0 + b01*x1 + b02*x2;
  v4f y1 = b10*x0 + b11*x1 + b12*x2;
  v4f y2 = b20*x0 + b21*x1 + b22*x2;

  __builtin_nontemporal_store(y0, (v4f*)(out + 0*T + o));
  __builtin_nontemporal_store(y1, (v4f*)(out + 1*T + o));
  __builtin_nontemporal_store(y2, (v4f*)(out + 2*T + o));

  v4f a0 = *(const v4f*)(X + 7*T + o);
  v4f a1 = *(const v4f*)(X + 8*T + o);
  v4f a2 = *(const v4f*)(X + 9*T + o);
  __builtin_nontemporal_store(a0, (v4f*)(out + 3*T + o));
  __builtin_nontemporal_store(a1, (v4f*)(out + 4*T + o));
  __builtin_nontemporal_store(a2, (v4f*)(out + 5*T + o));
}

extern "C" void kernel_launch(void* const* d_in, const int* in_sizes, int n_in,
                              void* d_out, int out_size, void* d_ws, size_t ws_size,
                              hipStream_t stream) {
  const float* X       = (const float*)d_in[0];
  const float* refAxis = (const float*)d_in[1];
  float* out = (float*)d_out;
  float* ws  = (float*)d_ws;

  zero_ws_kernel<<<1, 32, 0, stream>>>(ws);
  moments_kernel<<<512, 256, 0, stream>>>(X, ws);
  solve_kernel<<<1, 64, 0, stream>>>(X, refAxis, ws);

  const int n4 = T_LEN / 4;
  transform_kernel<<<(n4 + 255) / 256, 256, 0, stream>>>(X, ws + 16, out);
}